// GlobalAwareSimilarityLoss_69140383531481
// MI455X (gfx1250) — compile-verified
//
#include <hip/hip_runtime.h>
#include <hip/hip_bf16.h>
#include <math.h>

#define EPS_F 1e-12f

typedef __attribute__((ext_vector_type(2))) float v2f;
typedef __attribute__((ext_vector_type(8))) float v8f;

// ---------------------------------------------------------------------------
// Stage 1: per-pixel channel sum-of-squares for out & gt, write difference of
//          p = norm^2 / max(norm, eps).  x layout: [B, C, H, W] (f32).
//          Block = 512 threads = 8 channel-groups x 64 pixel-quads covering
//          256 pixels.  Each thread streams float4 (global_load_b128): 512 B
//          per wave per issue slot -> minimal issue overhead at 23.3 TB/s.
// ---------------------------------------------------------------------------
__global__ __launch_bounds__(512) void sumsq_diff_kernel(
    const float* __restrict__ xo, const float* __restrict__ xg,
    float* __restrict__ dOut, int C, int HW) {
  const int t  = threadIdx.x;
  const int q  = t & 63;              // pixel-quad index within block
  const int cg = t >> 6;              // 0..7 channel group
  const int Cq = C >> 3;              // channels per group

  const int pixBlock = blockIdx.x * 256;       // 256 pixels per block
  const int b        = pixBlock / HW;          // uniform per block
  const int pixInImg = (pixBlock % HW) + q * 4;
  const int strideQ  = HW >> 2;                // float4 stride per channel

  const size_t off = ((size_t)b * C + (size_t)cg * Cq) * (size_t)HW +
                     (size_t)pixInImg;
  const float4* po4 = reinterpret_cast<const float4*>(xo + off);
  const float4* pg4 = reinterpret_cast<const float4*>(xg + off);

  float4 so = {0.f, 0.f, 0.f, 0.f};
  float4 sg = {0.f, 0.f, 0.f, 0.f};
#pragma unroll 4
  for (int c = 0; c < Cq; ++c) {
    float4 a = po4[(size_t)c * strideQ];
    so.x = fmaf(a.x, a.x, so.x);
    so.y = fmaf(a.y, a.y, so.y);
    so.z = fmaf(a.z, a.z, so.z);
    so.w = fmaf(a.w, a.w, so.w);
    float4 g = pg4[(size_t)c * strideQ];
    sg.x = fmaf(g.x, g.x, sg.x);
    sg.y = fmaf(g.y, g.y, sg.y);
    sg.z = fmaf(g.z, g.z, sg.z);
    sg.w = fmaf(g.w, g.w, sg.w);
  }

  __shared__ float4 ro[512];
  __shared__ float4 rg[512];
  ro[t] = so;
  rg[t] = sg;
  __syncthreads();

  // Combine the 8 channel groups; one thread per output pixel (t < 256).
  if (t < 256) {
    const int pq = t >> 2;   // pixel-quad
    const int j  = t & 3;    // component within quad
    float SO = 0.f, SG = 0.f;
#pragma unroll
    for (int g = 0; g < 8; ++g) {
      SO += reinterpret_cast<const float*>(&ro[g * 64 + pq])[j];
      SG += reinterpret_cast<const float*>(&rg[g * 64 + pq])[j];
    }
    float no = sqrtf(SO);
    float ng = sqrtf(SG);
    float pO = (no * no) / fmaxf(no, EPS_F);
    float pG = (ng * ng) / fmaxf(ng, EPS_F);
    dOut[pixBlock + t] = pO - pG;
  }
}

// ---------------------------------------------------------------------------
// 16x16 output tile of X(NxN) * Y(NxN) via V_WMMA_F32_16X16X4_F32.
// ISA layouts (wave32):
//   A 16x4 : lanes 0-15 rows M=0..15; VGPR0 = K=(half*2), VGPR1 = K=(half*2)+1
//   B 4x16 : lanes hold N=lane&15; VGPR0 = K=(half*2), VGPR1 = K=(half*2)+1
//   C/D    : VGPR r, lanes 0-15 -> M=r, lanes 16-31 -> M=r+8, N=lane&15
// At == true reads X transposed (used for S^T without materializing it).
// ---------------------------------------------------------------------------
template <int N>
__device__ __forceinline__ v8f mm16_f32(const float* __restrict__ X,
                                        const float* __restrict__ Y,
                                        int rowBase, int colBase, bool At) {
  const int lane = threadIdx.x & 31;
  const int half = lane >> 4;   // 0 or 1
  const int l15  = lane & 15;
  v8f acc = {};
#pragma unroll
  for (int k0 = 0; k0 < N; k0 += 4) {
    const int ka = k0 + half * 2;
    v2f a, b;
    if (At) {
      a.x = X[(size_t)ka * N + rowBase + l15];
      a.y = X[(size_t)(ka + 1) * N + rowBase + l15];
    } else {
      a.x = X[(size_t)(rowBase + l15) * N + ka];
      a.y = X[(size_t)(rowBase + l15) * N + ka + 1];
    }
    b.x = Y[(size_t)ka * N + colBase + l15];
    b.y = Y[(size_t)(ka + 1) * N + colBase + l15];
    acc = __builtin_amdgcn_wmma_f32_16x16x4_f32(
        /*neg_a=*/false, a, /*neg_b=*/false, b,
        /*c_mod=*/(short)0, acc, /*reuse_a=*/false, /*reuse_b=*/false);
  }
  return acc;
}

// ---------------------------------------------------------------------------
// Stage 2: per-image box-sum losses for k in {3,5,7} via Gamma = S^T (D S),
// S = NxN band matrix of ones (columns >= N-k+1 zeroed).  One block per batch
// image; NW waves each own (N/16)^2 / NW output tiles.  Accumulates
// sum(Gamma[valid]^2) / (8*V*V) per k, deterministic tree reduce, writes one
// partial per block.
// ---------------------------------------------------------------------------
template <int N, int NW>
__global__ __launch_bounds__(NW * 32) void boxsum_loss_kernel(
    const float* __restrict__ d, float* __restrict__ lossPart, int partBase) {
  constexpr int NT      = NW * 32;
  constexpr int TILES   = (N / 16) * (N / 16);
  constexpr int TPW     = TILES / NW;  // tiles per wave
  constexpr int TCOLS   = N / 16;

  __shared__ float sD[N * N];
  __shared__ float sS[N * N];
  __shared__ float sR[N * N];

  const int t    = threadIdx.x;
  const int wave = t >> 5;
  const int lane = t & 31;
  const int half = lane >> 4;
  const int l15  = lane & 15;
  const int b    = blockIdx.x;

  for (int i = t; i < N * N; i += NT) sD[i] = d[(size_t)b * N * N + i];
  __syncthreads();

  float lossLocal = 0.0f;
  const int ks[3] = {3, 5, 7};

#pragma unroll
  for (int ki = 0; ki < 3; ++ki) {
    const int kk = ks[ki];
    const int V  = N - kk + 1;
    const float invCnt = 1.0f / (8.0f * (float)V * (float)V);

    // Build band summation matrix S[j][x'] = (x'<V && x'<=j<x'+k)
    for (int i = t; i < N * N; i += NT) {
      int j  = i / N;
      int xp = i - j * N;
      sS[i] = (xp < V && j >= xp && j < xp + kk) ? 1.0f : 0.0f;
    }
    __syncthreads();

    // R = D * S
#pragma unroll
    for (int tt = 0; tt < TPW; ++tt) {
      const int tile = wave * TPW + tt;
      const int ti = (tile / TCOLS) * 16;
      const int tj = (tile % TCOLS) * 16;
      v8f acc = mm16_f32<N>(sD, sS, ti, tj, /*At=*/false);
#pragma unroll
      for (int r = 0; r < 8; ++r) {
        const int m = ti + r + half * 8;
        const int n = tj + l15;
        sR[m * N + n] = acc[r];
      }
    }
    __syncthreads();

    // G = S^T * R ; accumulate squared values over valid region
#pragma unroll
    for (int tt = 0; tt < TPW; ++tt) {
      const int tile = wave * TPW + tt;
      const int ti = (tile / TCOLS) * 16;
      const int tj = (tile % TCOLS) * 16;
      v8f acc = mm16_f32<N>(sS, sR, ti, tj, /*At=*/true);
#pragma unroll
      for (int r = 0; r < 8; ++r) {
        const int m = ti + r + half * 8;
        const int n = tj + l15;
        float g = acc[r];
        if (m < V && n < V) lossLocal = fmaf(g * g, invCnt, lossLocal);
      }
    }
    __syncthreads();  // sS/sR consumed; safe to rebuild next k
  }

  // Deterministic block tree reduction (reuse sS; N*N >= NT).
  sS[t] = lossLocal;
  __syncthreads();
#pragma unroll
  for (int s = NT / 2; s > 0; s >>= 1) {
    if (t < s) sS[t] += sS[t + s];
    __syncthreads();
  }
  if (t == 0) lossPart[partBase + b] = sS[0];
}

// ---------------------------------------------------------------------------
// Stage 3: fixed-order final sum of 16 partials -> scalar loss.
// ---------------------------------------------------------------------------
__global__ void final_sum_kernel(const float* __restrict__ lossPart,
                                 float* __restrict__ out) {
  if (threadIdx.x == 0 && blockIdx.x == 0) {
    float s = 0.0f;
    for (int i = 0; i < 16; ++i) s += lossPart[i];
    out[0] = s;
  }
}

// ---------------------------------------------------------------------------
extern "C" void kernel_launch(void* const* d_in, const int* in_sizes, int n_in,
                              void* d_out, int out_size, void* d_ws,
                              size_t ws_size, hipStream_t stream) {
  (void)in_sizes; (void)n_in; (void)out_size; (void)ws_size;

  const float* of0 = (const float*)d_in[0];  // [8,256,64,64]
  const float* of1 = (const float*)d_in[1];  // [8,512,32,32]
  const float* gf0 = (const float*)d_in[2];
  const float* gf1 = (const float*)d_in[3];

  float* ws       = (float*)d_ws;
  float* d0       = ws;                   // 8*64*64   = 32768 floats
  float* d1       = ws + 32768;           // 8*32*32   =  8192 floats
  float* lossPart = ws + 32768 + 8192;    // 16 floats

  // Stage 1: bandwidth-bound channel reductions -> difference fields.
  //          256 pixels / block: feat0 -> 128 blocks, feat1 -> 32 blocks.
  sumsq_diff_kernel<<<128, 512, 0, stream>>>(of0, gf0, d0, 256, 64 * 64);
  sumsq_diff_kernel<<<32, 512, 0, stream>>>(of1, gf1, d1, 512, 32 * 32);

  // Stage 2: WMMA box-sum + masked MSE partials (8 blocks each).
  boxsum_loss_kernel<64, 8><<<8, 256, 0, stream>>>(d0, lossPart, 0);
  boxsum_loss_kernel<32, 4><<<8, 128, 0, stream>>>(d1, lossPart, 8);

  // Stage 3: scalar output.
  final_sum_kernel<<<1, 32, 0, stream>>>(lossPart, (float*)d_out);
}